// TensorDense_6725918785757
// MI455X (gfx1250) — compile-verified
//
#include <hip/hip_runtime.h>

// Fused e3nn-style tensor product for MI455X (gfx1250, wave32).
//
// Per node: s1=s@W1_0, v1=v@W1_1 (per xyz), s2,v2 likewise. All outputs are
// GEMMs over kappa=(i*32+j) with features f(i,j)=sum_t coef_t*A_t[i]*B_t[j]
// (rank<=3), A_t in {s1,v1c}, B_t in {s2,v2c}. Real CG coefficients:
//   l=0: -1/sqrt(3) * delta_ab
//   l=1: -eps_mab / sqrt(2)
//   l=2: symmetric traceless (1/sqrt2 off-diag, diag(-1,2,-1)/sqrt6, diag(-1,0,1)/sqrt2)
// Stage-3 = 4.3G f32 MACs mapped 100% to v_wmma_f32_16x16x4_f32.
// All piece metadata is constexpr/templated so the hot loops contain ONLY:
// ds_load_b64 (A-term pairs, conflict-free via stride-34 LDS padding),
// global_load_b32 with SGPR weight base (L2-resident), v_fma, v_wmma.

typedef float v2f __attribute__((ext_vector_type(2)));
typedef float v8f __attribute__((ext_vector_type(8)));

#define NODES_PER_WG 32
#define FSTRIDE 34
#define OFF_S1 0
#define OFF_S2 (32 * FSTRIDE)
#define OFF_V1 (2 * 32 * FSTRIDE)
#define OFF_V2 (5 * 32 * FSTRIDE)
#define FEAT_TOT (8 * 32 * FSTRIDE)

#define S3 0.57735026918962576f /* 1/sqrt(3) */
#define R2 0.70710678118654752f /* 1/sqrt(2) */
#define R6 0.40824829046386302f /* 1/sqrt(6) */

// ---- compile-time piece tables ---------------------------------------------
// Output groups (32 cols each): 0:out0 | 1-3:out1o c | 4-6:out1e m | 7-11:out2e m
constexpr int PSTART[12] = {0, 2, 4, 6, 8, 9, 10, 11, 12, 13, 14, 15};
constexpr int PCNT[12]   = {2, 2, 2, 2, 1, 1, 1, 1, 1, 1, 1, 1};
// wsel: 0=W3_0e 1=W3_1o 2=W3_1e 3=W3_2e ; selectors: 0=scalar, 1..3=vector comp
constexpr int PS_WSEL[16]  = {0, 0, 1, 1, 1, 1, 1, 1, 2, 2, 2, 3, 3, 3, 3, 3};
constexpr int PS_WBASE[16] = {0, 1024, 0, 1024, 0, 1024, 0, 1024, 0, 0, 0, 0, 0, 0, 0, 0};
constexpr int PS_NT[16]    = {1, 3, 1, 1, 1, 1, 1, 1, 2, 2, 2, 2, 2, 3, 2, 2};
constexpr int PS_A[16][3] = {
    {0,0,0}, {1,2,3}, {0,0,0}, {1,0,0}, {0,0,0}, {2,0,0}, {0,0,0}, {3,0,0},
    {3,2,0}, {1,3,0}, {2,1,0}, {1,3,0}, {1,2,0}, {1,2,3}, {2,3,0}, {1,3,0}};
constexpr int PS_B[16][3] = {
    {0,0,0}, {1,2,3}, {1,0,0}, {0,0,0}, {2,0,0}, {0,0,0}, {3,0,0}, {0,0,0},
    {2,3,0}, {3,1,0}, {1,2,0}, {3,1,0}, {2,1,0}, {1,2,3}, {3,2,0}, {1,3,0}};
constexpr float PS_C[16][3] = {
    {1.f,0.f,0.f}, {-S3,-S3,-S3}, {1.f,0.f,0.f}, {1.f,0.f,0.f},
    {1.f,0.f,0.f}, {1.f,0.f,0.f}, {1.f,0.f,0.f}, {1.f,0.f,0.f},
    {R2,-R2,0.f}, {R2,-R2,0.f}, {R2,-R2,0.f},
    {R2,R2,0.f}, {R2,R2,0.f}, {-R6,2.f*R6,-R6}, {R2,R2,0.f}, {-R2,R2,0.f}};

constexpr int featAOff(int sel) { return sel == 0 ? OFF_S1 : OFF_V1 + (sel - 1) * 32 * FSTRIDE; }
constexpr int featBOff(int sel) { return sel == 0 ? OFF_S2 : OFF_V2 + (sel - 1) * 32 * FSTRIDE; }

struct Ctx {
  const float* __restrict__ W0e;
  const float* __restrict__ W1o;
  const float* __restrict__ W1e;
  const float* __restrict__ W2e;
  const float* featL;  // LDS
  float* __restrict__ out;
  int nodeBase;
  int n0;  // lane & 15 : A row / B-D column
  int hi;  // lane >> 4 : K-pair select
};

template <int PI>
__device__ __forceinline__ void run_piece(const Ctx& c, int bcol, v8f& acc0, v8f& acc1) {
  constexpr int wsel  = PS_WSEL[PI];
  constexpr int wbase = PS_WBASE[PI];
  constexpr int NT    = PS_NT[PI];
  constexpr int aO0 = featAOff(PS_A[PI][0]), aO1 = featAOff(PS_A[PI][1]), aO2 = featAOff(PS_A[PI][2]);
  constexpr int bO0 = featBOff(PS_B[PI][0]), bO1 = featBOff(PS_B[PI][1]), bO2 = featBOff(PS_B[PI][2]);
  constexpr float C0 = PS_C[PI][0], C1 = PS_C[PI][1], C2 = PS_C[PI][2];

  const float* __restrict__ W =
      (wsel == 0) ? c.W0e : (wsel == 1) ? c.W1o : (wsel == 2) ? c.W1e : c.W2e;
  const float* featL = c.featL;

  const int rowA0 = c.n0 * FSTRIDE;
  const int rowA1 = (c.n0 + 16) * FSTRIDE;
  const int laneW = bcol + (c.hi << 6);  // (2*hi)*32 + bcol, loop-invariant VGPR part

  for (int i = 0; i < 32; ++i) {
    const float A00 = C0 * featL[aO0 + rowA0 + i];
    const float A01 = C0 * featL[aO0 + rowA1 + i];
    float A10 = 0.f, A11 = 0.f, A20 = 0.f, A21 = 0.f;
    if constexpr (NT >= 2) {
      A10 = C1 * featL[aO1 + rowA0 + i];
      A11 = C1 * featL[aO1 + rowA1 + i];
    }
    if constexpr (NT >= 3) {
      A20 = C2 * featL[aO2 + rowA0 + i];
      A21 = C2 * featL[aO2 + rowA1 + i];
    }
    const int wrow = ((wbase + (i << 5)) << 5) + laneW;
#pragma unroll
    for (int js = 0; js < 8; ++js) {
      const int j0 = (js << 2) + (c.hi << 1);
      v2f a0, a1;
      {
        const v2f b0 = *(const v2f*)&featL[bO0 + rowA0 + j0];
        const v2f b1 = *(const v2f*)&featL[bO0 + rowA1 + j0];
        a0 = A00 * b0;
        a1 = A01 * b1;
      }
      if constexpr (NT >= 2) {
        const v2f b0 = *(const v2f*)&featL[bO1 + rowA0 + j0];
        const v2f b1 = *(const v2f*)&featL[bO1 + rowA1 + j0];
        a0 += A10 * b0;
        a1 += A11 * b1;
      }
      if constexpr (NT >= 3) {
        const v2f b0 = *(const v2f*)&featL[bO2 + rowA0 + j0];
        const v2f b1 = *(const v2f*)&featL[bO2 + rowA1 + j0];
        a0 += A20 * b0;
        a1 += A21 * b1;
      }
      v2f bf;
      bf.x = W[wrow + (js << 7)];
      bf.y = W[wrow + (js << 7) + 32];
      acc0 = __builtin_amdgcn_wmma_f32_16x16x4_f32(false, a0, false, bf,
                                                   (short)0, acc0, false, false);
      acc1 = __builtin_amdgcn_wmma_f32_16x16x4_f32(false, a1, false, bf,
                                                   (short)0, acc1, false, false);
    }
  }
}

template <int G>
__device__ __forceinline__ void run_unit(const Ctx& c, int ntile) {
  const int bcol = c.n0 + (ntile << 4);
  v8f acc0 = {0.f,0.f,0.f,0.f,0.f,0.f,0.f,0.f};
  v8f acc1 = {0.f,0.f,0.f,0.f,0.f,0.f,0.f,0.f};
  run_piece<PSTART[G]>(c, bcol, acc0, acc1);
  if constexpr (PCNT[G] == 2) run_piece<PSTART[G] + 1>(c, bcol, acc0, acc1);

  // C/D layout: VGPR r -> lanes 0-15: M=r, lanes 16-31: M=r+8; N = lane&15
  const int k = bcol;
  int col;
  if constexpr (G == 0)      col = k;                      // out0
  else if constexpr (G < 4)  col = 32  + k * 3 + (G - 1);  // out1o (k*3+c)
  else if constexpr (G < 7)  col = 128 + k * 3 + (G - 4);  // out1e (k*3+m)
  else                       col = 224 + k * 5 + (G - 7);  // out2e (k*5+m)
#pragma unroll
  for (int r = 0; r < 8; ++r) {
    const int m = r + (c.hi << 3);
    c.out[(size_t)(c.nodeBase + m) * 384 + col]      = acc0[r];
    c.out[(size_t)(c.nodeBase + 16 + m) * 384 + col] = acc1[r];
  }
}

__global__ __launch_bounds__(256, 2)
void fused_tp_kernel(const float* __restrict__ x,
                     const float* __restrict__ W1_0, const float* __restrict__ W1_1,
                     const float* __restrict__ W2_0, const float* __restrict__ W2_1,
                     const float* __restrict__ W3_0e, const float* __restrict__ W3_1o,
                     const float* __restrict__ W3_1e, const float* __restrict__ W3_2e,
                     float* __restrict__ out) {
  __shared__ float xS[NODES_PER_WG * 132];  // x tile, 16B-aligned rows
  __shared__ float wS[4 * 1024];            // W1_0 | W1_1 | W2_0 | W2_1
  __shared__ float featL[FEAT_TOT];         // s1 | s2 | v1[3] | v2[3], stride 34

  const int tid = threadIdx.x;
  const int nodeBase = blockIdx.x * NODES_PER_WG;

  // ---- Phase A0: stage x tile (32x128) and stage-1 weights into LDS ----
  for (int idx = tid; idx < NODES_PER_WG * 32; idx += 256) {
    const int n = idx >> 5, q = idx & 31;
    const float4 v = *(const float4*)(x + (size_t)(nodeBase + n) * 128 + q * 4);
    *(float4*)&xS[n * 132 + q * 4] = v;
  }
  for (int idx = tid; idx < 1024; idx += 256) {
    wS[idx]        = W1_0[idx];
    wS[1024 + idx] = W1_1[idx];
    wS[2048 + idx] = W2_0[idx];
    wS[3072 + idx] = W2_1[idx];
  }
  __syncthreads();

  // ---- Phase A1: stage-1 linears (VALU; ~1.5% of MACs). Thread = (node, 4 k's) ----
  {
    const int n  = tid >> 3;
    const int k0 = (tid & 7) * 4;
    float s1a[4] = {0.f,0.f,0.f,0.f}, s2a[4] = {0.f,0.f,0.f,0.f};
    float v1a[3][4] = {{0}}, v2a[3][4] = {{0}};
    for (int m = 0; m < 32; ++m) {
      const float xs  = xS[n * 132 + m];
      const float xv0 = xS[n * 132 + 32 + m * 3 + 0];
      const float xv1 = xS[n * 132 + 32 + m * 3 + 1];
      const float xv2 = xS[n * 132 + 32 + m * 3 + 2];
      float w10[4], w11[4], w20[4], w21[4];
      *(float4*)w10 = *(const float4*)&wS[m * 32 + k0];
      *(float4*)w11 = *(const float4*)&wS[1024 + m * 32 + k0];
      *(float4*)w20 = *(const float4*)&wS[2048 + m * 32 + k0];
      *(float4*)w21 = *(const float4*)&wS[3072 + m * 32 + k0];
#pragma unroll
      for (int q = 0; q < 4; ++q) {
        s1a[q] += xs * w10[q];
        s2a[q] += xs * w20[q];
        v1a[0][q] += xv0 * w11[q]; v1a[1][q] += xv1 * w11[q]; v1a[2][q] += xv2 * w11[q];
        v2a[0][q] += xv0 * w21[q]; v2a[1][q] += xv1 * w21[q]; v2a[2][q] += xv2 * w21[q];
      }
    }
#pragma unroll
    for (int q = 0; q < 4; ++q) {
      featL[OFF_S1 + n * FSTRIDE + k0 + q] = s1a[q];
      featL[OFF_S2 + n * FSTRIDE + k0 + q] = s2a[q];
#pragma unroll
      for (int cc = 0; cc < 3; ++cc) {
        featL[OFF_V1 + cc * 32 * FSTRIDE + n * FSTRIDE + k0 + q] = v1a[cc][q];
        featL[OFF_V2 + cc * 32 * FSTRIDE + n * FSTRIDE + k0 + q] = v2a[cc][q];
      }
    }
  }
  __syncthreads();

  // ---- Phase B: WMMA GEMMs (M=2x16 nodes, N=16 cols, K=4 per instruction) ----
  const int lane = tid & 31;
  const int wave = tid >> 5;

  Ctx c;
  c.W0e = W3_0e; c.W1o = W3_1o; c.W1e = W3_1e; c.W2e = W3_2e;
  c.featL = featL;
  c.out = out;
  c.nodeBase = nodeBase;
  c.n0 = lane & 15;
  c.hi = lane >> 4;

  // Wave -> unit assignment (uniform switch); every wave executes 2048 WMMAs.
  switch (wave) {
    case 0: run_unit<0>(c, 0);  run_unit<0>(c, 1);  break;
    case 1: run_unit<1>(c, 0);  run_unit<1>(c, 1);  break;
    case 2: run_unit<2>(c, 0);  run_unit<2>(c, 1);  break;
    case 3: run_unit<3>(c, 0);  run_unit<3>(c, 1);  break;
    case 4: run_unit<4>(c, 0);  run_unit<4>(c, 1);  run_unit<5>(c, 0);  run_unit<5>(c, 1);  break;
    case 5: run_unit<6>(c, 0);  run_unit<6>(c, 1);  run_unit<7>(c, 0);  run_unit<7>(c, 1);  break;
    case 6: run_unit<8>(c, 0);  run_unit<8>(c, 1);  run_unit<9>(c, 0);  run_unit<9>(c, 1);  break;
    default: run_unit<10>(c, 0); run_unit<10>(c, 1); run_unit<11>(c, 0); run_unit<11>(c, 1); break;
  }
}

extern "C" void kernel_launch(void* const* d_in, const int* in_sizes, int n_in,
                              void* d_out, int out_size, void* d_ws, size_t ws_size,
                              hipStream_t stream) {
  (void)n_in; (void)out_size; (void)d_ws; (void)ws_size;
  const float* x     = (const float*)d_in[0];
  const float* W1_0  = (const float*)d_in[1];
  const float* W1_1  = (const float*)d_in[2];
  const float* W2_0  = (const float*)d_in[3];
  const float* W2_1  = (const float*)d_in[4];
  const float* W3_0e = (const float*)d_in[5];
  const float* W3_1o = (const float*)d_in[6];
  const float* W3_1e = (const float*)d_in[7];
  const float* W3_2e = (const float*)d_in[8];
  float* outp = (float*)d_out;

  const int N = in_sizes[0] / 128;      // 8192
  const int grid = N / NODES_PER_WG;    // 256 workgroups of 32 nodes
  fused_tp_kernel<<<grid, 256, 0, stream>>>(x, W1_0, W1_1, W2_0, W2_1,
                                            W3_0e, W3_1o, W3_1e, W3_2e, outp);
}